// O_SVGD_42700564857286
// MI455X (gfx1250) — compile-verified
//
#include <hip/hip_runtime.h>
#include <hip/hip_bf16.h>
#include <math.h>

#define N_PART 2048
#define DIM 64
#define NN (N_PART * N_PART)   // 4194304
#define K0 2097151u            // 0-indexed lower-middle rank (NN/2 - 1)

typedef float v2f __attribute__((ext_vector_type(2)));
typedef float v8f __attribute__((ext_vector_type(8)));

struct MState {
  unsigned prefix;    // accumulated radix-select prefix
  unsigned kRemain;   // remaining rank within current prefix
  unsigned aBits;     // selected value a (uint bits of float)
  unsigned cntLE;     // count of elements <= a
  unsigned minAbove;  // min uint value strictly > a
  float hh;
  float invhh;        // 1/hh
  float beta;         // 2/hh
};

// ---------------- per-row stats: r = ||x||^2, alpha = (x.mu + 63)/r ----------------
__global__ void row_stats(const float* __restrict__ X, const float* __restrict__ mu,
                          float* __restrict__ r, float* __restrict__ rinv,
                          float* __restrict__ alpha) {
  __shared__ float s1[64];
  __shared__ float s2[64];
  int row = blockIdx.x, t = threadIdx.x;
  float xv = X[row * DIM + t];
  s1[t] = xv * xv;
  s2[t] = xv * mu[t];
  __syncthreads();
  for (int off = 32; off > 0; off >>= 1) {
    if (t < off) { s1[t] += s1[t + off]; s2[t] += s2[t + off]; }
    __syncthreads();
  }
  if (t == 0) {
    float rr = s1[0];
    float ir = 1.0f / rr;
    r[row] = rr;
    rinv[row] = ir;
    alpha[row] = (s2[0] + (float)(DIM - 1)) * ir;
  }
}

// ---------------- Gram matrix dot = X X^T via fp32 WMMA 16x16x4 ----------------
__global__ void __launch_bounds__(256) gram_wmma(const float* __restrict__ X,
                                                 float* __restrict__ dotm) {
  int wave = blockIdx.x * 8 + (threadIdx.x >> 5);  // 16384 waves = 128x128 tiles
  int lane = threadIdx.x & 31;
  int tileI = wave >> 7;
  int tileJ = wave & 127;
  int lr = lane & 15;
  int hi = lane >> 4;  // 0: K=0,1  1: K=2,3 within each 4-slab
  const float* arow = X + (size_t)(tileI * 16 + lr) * DIM;
  const float* brow = X + (size_t)(tileJ * 16 + lr) * DIM;
  v8f c = {0.f, 0.f, 0.f, 0.f, 0.f, 0.f, 0.f, 0.f};
#pragma unroll
  for (int kk = 0; kk < DIM; kk += 4) {
    v2f a, b;
    a.x = arow[kk + 2 * hi];
    a.y = arow[kk + 2 * hi + 1];
    b.x = brow[kk + 2 * hi];
    b.y = brow[kk + 2 * hi + 1];
    c = __builtin_amdgcn_wmma_f32_16x16x4_f32(false, a, false, b, (short)0, c,
                                              false, false);
  }
  // C layout: VGPR v -> M = v + 8*hi, N = lr
  float* out = dotm + (size_t)(tileI * 16 + 8 * hi) * N_PART + tileJ * 16 + lr;
#pragma unroll
  for (int v = 0; v < 8; ++v) out[(size_t)v * N_PART] = c[v];
}

// ---------------- generic zero ----------------
__global__ void zero_f32(float* p, int count) {
  int i = blockIdx.x * blockDim.x + threadIdx.x;
  if (i < count) p[i] = 0.0f;
}
__global__ void zero_u32(unsigned* p, int count) {
  int i = blockIdx.x * blockDim.x + threadIdx.x;
  if (i < count) p[i] = 0u;
}
__global__ void init_state(MState* st) {
  st->prefix = 0u;
  st->kRemain = K0;
  st->aBits = 0u;
  st->cntLE = 0u;
  st->minAbove = 0x7fffffffu;
  st->hh = 1.0f; st->invhh = 1.0f; st->beta = 2.0f;
}

// ---------------- pd on the fly: pd_ij = max(r_i + r_j - 2 dot_ij, 0) ----------------
__device__ __forceinline__ unsigned pd_bits(const float* dotm, const float* r, size_t idx) {
  int i = (int)(idx >> 11);
  int j = (int)(idx & (N_PART - 1));
  float pd = fmaxf(r[i] + r[j] - 2.0f * dotm[idx], 0.0f);
  return __float_as_uint(pd);  // nonneg float: uint order == float order
}

// histogram radix passes: level0 bits[30:20] (2048 bins), level1 bits[19:9] (2048), level2 bits[8:0] (512)
__global__ void __launch_bounds__(256) hist_pass(const float* __restrict__ dotm,
                                                 const float* __restrict__ r,
                                                 unsigned* __restrict__ histo,
                                                 const MState* __restrict__ st, int level) {
  __shared__ unsigned lh[2048];
  for (int i = threadIdx.x; i < 2048; i += blockDim.x) lh[i] = 0u;
  __syncthreads();
  unsigned prefix = st->prefix;
  size_t stride = (size_t)gridDim.x * blockDim.x;
  for (size_t idx = (size_t)blockIdx.x * blockDim.x + threadIdx.x; idx < (size_t)NN;
       idx += stride) {
    unsigned u = pd_bits(dotm, r, idx);
    if (level == 0) {
      atomicAdd(&lh[u >> 20], 1u);
    } else if (level == 1) {
      if ((u >> 20) == prefix) atomicAdd(&lh[(u >> 9) & 2047u], 1u);
    } else {
      if ((u >> 9) == prefix) atomicAdd(&lh[u & 511u], 1u);
    }
  }
  __syncthreads();
  for (int i = threadIdx.x; i < 2048; i += blockDim.x)
    if (lh[i]) atomicAdd(&histo[i], lh[i]);
}

__global__ void select_pass(unsigned* histo, MState* st, int level) {
  unsigned k = st->kRemain;
  int nb = (level == 2) ? 512 : 2048;
  unsigned cum = 0u, before = 0u;
  int bin = nb - 1;
  for (int b = 0; b < nb; ++b) {
    unsigned c = histo[b];
    if (k < cum + c) { bin = b; before = cum; break; }
    cum += c;
  }
  st->kRemain = k - before;
  if (level == 0) st->prefix = (unsigned)bin;
  else if (level == 1) st->prefix = (st->prefix << 11) | (unsigned)bin;
  else {
    st->prefix = (st->prefix << 9) | (unsigned)bin;
    st->aBits = st->prefix;
    st->cntLE = 0u;
    st->minAbove = 0x7fffffffu;
  }
  for (int b = 0; b < 2048; ++b) histo[b] = 0u;  // reset for next pass
}

__global__ void __launch_bounds__(256) lemin_pass(const float* __restrict__ dotm,
                                                  const float* __restrict__ r, MState* st) {
  __shared__ unsigned scnt[256];
  __shared__ unsigned smin[256];
  unsigned a = st->aBits;
  unsigned lc = 0u, lm = 0x7fffffffu;
  size_t stride = (size_t)gridDim.x * blockDim.x;
  for (size_t idx = (size_t)blockIdx.x * blockDim.x + threadIdx.x; idx < (size_t)NN;
       idx += stride) {
    unsigned u = pd_bits(dotm, r, idx);
    if (u <= a) lc++;
    else lm = (u < lm) ? u : lm;
  }
  int t = threadIdx.x;
  scnt[t] = lc; smin[t] = lm;
  __syncthreads();
  for (int off = 128; off > 0; off >>= 1) {
    if (t < off) {
      scnt[t] += scnt[t + off];
      smin[t] = (smin[t + off] < smin[t]) ? smin[t + off] : smin[t];
    }
    __syncthreads();
  }
  if (t == 0) {
    atomicAdd(&st->cntLE, scnt[0]);
    atomicMin(&st->minAbove, smin[0]);
  }
}

__global__ void finalize_state(MState* st) {
  unsigned a = st->aBits;
  // element at 0-index NN/2 equals a iff cntLE >= NN/2 + 1
  unsigned b = (st->cntLE >= (K0 + 2u)) ? a : st->minAbove;
  if (b == 0x7fffffffu) b = a;
  float med = 0.5f * (__uint_as_float(a) + __uint_as_float(b));
  float h = med / 7.6246189861593985f;  // log(2048)
  float hh = h + 1e-6f;
  st->hh = hh;
  st->invhh = 1.0f / hh;
  st->beta = 2.0f / hh;
}

// ---------------- W = M @ X with M fused (exp kernel + coefficients), plus K1 ----------------
#define NCHUNK 16
#define CHUNK (N_PART / NCHUNK)  // 128
__global__ void __launch_bounds__(256) mx_wmma(const float* __restrict__ X,
                                               const float* __restrict__ dotm,
                                               const float* __restrict__ r,
                                               const float* __restrict__ rinv,
                                               const float* __restrict__ alpha,
                                               const MState* __restrict__ st,
                                               float* __restrict__ W,
                                               float* __restrict__ K1) {
  int wave = blockIdx.x * 8 + (threadIdx.x >> 5);  // 2048 waves = 128 row-tiles x 16 chunks
  int lane = threadIdx.x & 31;
  int tile = wave >> 4;
  int chunk = wave & 15;
  int lr = lane & 15;
  int hi = lane >> 4;
  int rowA = tile * 16 + lr;
  float ri = r[rowA];
  float invhh = st->invhh;
  float beta = st->beta;
  const float* drow = dotm + (size_t)rowA * N_PART;
  v8f acc0 = {0.f,0.f,0.f,0.f,0.f,0.f,0.f,0.f};
  v8f acc1 = acc0, acc2 = acc0, acc3 = acc0;
  float k1p = 0.0f;
  int jbeg = chunk * CHUNK, jend = jbeg + CHUNK;
  for (int j0 = jbeg; j0 < jend; j0 += 4) {
    int jA = j0 + 2 * hi;
    v2f a;
#pragma unroll
    for (int q = 0; q < 2; ++q) {
      float dv = drow[jA + q];
      float rj = r[jA + q];
      float pd = fmaxf(ri + rj - 2.0f * dv, 0.0f);
      float kv = __expf(-pd * invhh);
      k1p += kv;
      a[q] = kv * (alpha[jA + q] + beta * dv * rinv[jA + q]);
    }
    const float* x0 = X + (size_t)jA * DIM + lr;
    const float* x1 = x0 + DIM;
    v2f b;
    b.x = x0[0];  b.y = x1[0];
    acc0 = __builtin_amdgcn_wmma_f32_16x16x4_f32(false, a, false, b, (short)0, acc0, false, false);
    b.x = x0[16]; b.y = x1[16];
    acc1 = __builtin_amdgcn_wmma_f32_16x16x4_f32(false, a, false, b, (short)0, acc1, false, false);
    b.x = x0[32]; b.y = x1[32];
    acc2 = __builtin_amdgcn_wmma_f32_16x16x4_f32(false, a, false, b, (short)0, acc2, false, false);
    b.x = x0[48]; b.y = x1[48];
    acc3 = __builtin_amdgcn_wmma_f32_16x16x4_f32(false, a, false, b, (short)0, acc3, false, false);
  }
  int outRow = tile * 16 + 8 * hi;
#pragma unroll
  for (int v = 0; v < 8; ++v) {
    atomicAdd(&W[(size_t)(outRow + v) * DIM +  0 + lr], acc0[v]);
    atomicAdd(&W[(size_t)(outRow + v) * DIM + 16 + lr], acc1[v]);
    atomicAdd(&W[(size_t)(outRow + v) * DIM + 32 + lr], acc2[v]);
    atomicAdd(&W[(size_t)(outRow + v) * DIM + 48 + lr], acc3[v]);
  }
  atomicAdd(&K1[rowA], k1p);
}

// ---------------- finish: v = K1*(mu + beta*x) - W; project; step ----------------
__global__ void finish(const float* __restrict__ X, const float* __restrict__ mu,
                       const float* __restrict__ W, const float* __restrict__ K1,
                       const float* __restrict__ r, const MState* __restrict__ st,
                       float* __restrict__ out) {
  __shared__ float sd[64];
  int row = blockIdx.x, t = threadIdx.x;
  float beta = st->beta;
  float x = X[row * DIM + t];
  float v = K1[row] * (mu[t] + beta * x) - W[row * DIM + t];
  sd[t] = x * v;
  __syncthreads();
  for (int off = 32; off > 0; off >>= 1) {
    if (t < off) sd[t] += sd[t + off];
    __syncthreads();
  }
  float p = sd[0];
  float g = v - x * ((p + (float)(DIM - 1)) / r[row]);
  float dx = g * (0.1f / (float)N_PART);
  dx = fminf(fmaxf(dx, -1000.0f), 1000.0f);
  out[row * DIM + t] = x + dx;
}

extern "C" void kernel_launch(void* const* d_in, const int* in_sizes, int n_in,
                              void* d_out, int out_size, void* d_ws, size_t ws_size,
                              hipStream_t stream) {
  (void)in_sizes; (void)n_in; (void)out_size; (void)ws_size;
  const float* X  = (const float*)d_in[0];   // [2048, 64]
  const float* mu = (const float*)d_in[1];   // [64]
  float* out = (float*)d_out;                // [2048, 64]

  char* w = (char*)d_ws;
  float* dotm  = (float*)w;                              // 2048*2048 f32 = 16 MB
  float* r     = (float*)(w + (size_t)NN * 4);           // 2048
  float* rinv  = r + N_PART;                             // 2048
  float* alpha = rinv + N_PART;                          // 2048
  float* K1    = alpha + N_PART;                         // 2048
  float* W     = K1 + N_PART;                            // 2048*64
  unsigned* histo = (unsigned*)(W + N_PART * DIM);       // 2048
  MState* st   = (MState*)(histo + 2048);

  row_stats<<<N_PART, DIM, 0, stream>>>(X, mu, r, rinv, alpha);
  gram_wmma<<<2048, 256, 0, stream>>>(X, dotm);

  zero_f32<<<(N_PART + N_PART * DIM + 255) / 256, 256, 0, stream>>>(K1, N_PART + N_PART * DIM);
  zero_u32<<<8, 256, 0, stream>>>(histo, 2048);
  init_state<<<1, 1, 0, stream>>>(st);

  hist_pass<<<2048, 256, 0, stream>>>(dotm, r, histo, st, 0);
  select_pass<<<1, 1, 0, stream>>>(histo, st, 0);
  hist_pass<<<2048, 256, 0, stream>>>(dotm, r, histo, st, 1);
  select_pass<<<1, 1, 0, stream>>>(histo, st, 1);
  hist_pass<<<2048, 256, 0, stream>>>(dotm, r, histo, st, 2);
  select_pass<<<1, 1, 0, stream>>>(histo, st, 2);
  lemin_pass<<<2048, 256, 0, stream>>>(dotm, r, st);
  finalize_state<<<1, 1, 0, stream>>>(st);

  mx_wmma<<<256, 256, 0, stream>>>(X, dotm, r, rinv, alpha, st, W, K1);
  finish<<<N_PART, DIM, 0, stream>>>(X, mu, W, K1, r, st, out);
}